// MoCo_12223476925100
// MI455X (gfx1250) — compile-verified
//
#include <hip/hip_runtime.h>
#include <math.h>

// ---------------- problem dims ----------------
#define D_N   768
#define D_B   12
#define D_F   64
#define D_EF  32
#define D_AC  16
#define D_EC  5
#define D_NH  10
#define D_HD  16
#define D_IN  160
#define D_NE  24576
#define D_NL  5

typedef __attribute__((ext_vector_type(2))) float v2f;
typedef __attribute__((ext_vector_type(8))) float v8f;

__device__ __forceinline__ v8f wmma4(v2f a, v2f b, v8f c) {
  // D = A(16x4 f32) * B(4x16 f32) + C(16x16 f32)
  return __builtin_amdgcn_wmma_f32_16x16x4_f32(false, a, false, b, (short)0, c, false, false);
}
__device__ __forceinline__ float silu_f(float x){ return x / (1.f + __expf(-x)); }

// ---------------- tiny utility kernels ----------------
__global__ void k_zero(float* p, int n){ int i = blockIdx.x*256 + threadIdx.x; if (i < n) p[i] = 0.f; }
__global__ void k_copy(float* d, const float* s, int n){ int i = blockIdx.x*256 + threadIdx.x; if (i < n) d[i] = s[i]; }

// ---------------- timestep embedding + AdaLN scale/shift per graph ----------------
__global__ void k_time(const float* t, const float* W1, const float* b1,
                       const float* W2, const float* b2,
                       const float* ASW, const float* ASb,
                       const float* AHW, const float* AHb,
                       float* scale_g, float* shift_g) {
  __shared__ float te0[64], h1[64], te[64];
  int g = blockIdx.x, j = threadIdx.x;
  float tv = t[g];
  int jm = j & 31;
  float fr = __expf(-logf(10000.f) * (float)jm / 32.f);
  te0[j] = (j < 32) ? __cosf(tv * fr) : __sinf(tv * fr);
  __syncthreads();
  float s = b1[j];
  for (int k2 = 0; k2 < 64; ++k2) s += te0[k2] * W1[k2*64 + j];
  h1[j] = silu_f(s);
  __syncthreads();
  s = b2[j];
  for (int k2 = 0; k2 < 64; ++k2) s += h1[k2] * W2[k2*64 + j];
  te[j] = s;
  __syncthreads();
  float sa = ASb[j], sh = AHb[j];
  for (int k2 = 0; k2 < 64; ++k2) { sa += te[k2]*ASW[k2*64+j]; sh += te[k2]*AHW[k2*64+j]; }
  scale_g[g*64 + j] = sa;
  shift_g[g*64 + j] = sh;
}

// ---------------- edge embedding: E = Ein @ W + b ----------------
__global__ void k_embed_e(const float* Ein, const float* W, const float* b, float* E) {
  int i = blockIdx.x*256 + threadIdx.x;
  if (i >= D_NE * D_EF) return;
  int e = i >> 5, c = i & 31;
  float s = b[c];
  #pragma unroll
  for (int k2 = 0; k2 < D_EC; ++k2) s += Ein[e*D_EC + k2] * W[k2*D_EF + c];
  E[i] = s;
}

// ---------------- atom embedding + LayerNorm + AdaLN ----------------
__global__ void k_embed_h(const float* Hin, const float* AW, const float* Ab,
                          const int* batch, const float* scale_g, const float* shift_g,
                          float* H) {
  __shared__ float r1[64], r2[64];
  int n = blockIdx.x, f = threadIdx.x;
  float h0 = Ab[f];
  #pragma unroll
  for (int a = 0; a < D_AC; ++a) h0 += Hin[n*D_AC + a] * AW[a*D_F + f];
  r1[f] = h0; r2[f] = h0*h0;
  __syncthreads();
  for (int s2 = 32; s2 > 0; s2 >>= 1) {
    if (f < s2) { r1[f] += r1[f+s2]; r2[f] += r2[f+s2]; }
    __syncthreads();
  }
  float mean = r1[0] / 64.f;
  float var  = r2[0] / 64.f - mean*mean;
  int g = batch[n];
  float xn = (h0 - mean) * rsqrtf(var + 1e-5f);
  H[n*D_F + f] = xn * (1.f + scale_g[g*D_F + f]) + shift_g[g*D_F + f];
}

// ---------------- per-dst edge counts (for segment_mean) ----------------
__global__ void k_counts(const int* Eidx, float* cnt) {
  int i = blockIdx.x*256 + threadIdx.x;
  if (i < D_NE) atomicAdd(&cnt[Eidx[D_NE + i]], 1.f);
}

// ---------------- generic WMMA GEMM: Out = act(A@B + bias [+ bias_scalar]) + beta*Cin ----------------
// one wave per 16x16 output tile; M must be a multiple of 16; K a multiple of 4.
// Out-of-range N lanes: B column is CLAMPED (unconditional load, no exec juggling);
// the garbage accumulator entries belong to unstored columns only.
__global__ __launch_bounds__(32) void k_gemm(
    const float* __restrict__ A, int lda,
    const float* __restrict__ B, int ldb,
    const float* __restrict__ bias, const float* __restrict__ bias_scalar,
    const float* __restrict__ Cin, int ldci,
    float* __restrict__ Out, int ldo,
    int Nc, int K, int ntN, int act, float beta) {
  int lane = threadIdx.x, l16 = lane & 15, hi = lane >> 4, koff = hi * 2;
  int m0 = (blockIdx.x / ntN) * 16;
  int n0 = (blockIdx.x % ntN) * 16;
  int arow = m0 + l16, bcol = n0 + l16;
  bool cok = (bcol < Nc);
  int bcolc = cok ? bcol : (Nc - 1);      // clamp: always a valid address
  const float* Ap = A + arow*lda;
  const float* Bp = B + bcolc;
  v8f acc = {};
  #pragma unroll 4
  for (int f0 = 0; f0 < K; f0 += 4) {
    int fa = f0 + koff;
    v2f a, b;
    a.x = Ap[fa];
    a.y = Ap[fa + 1];
    b.x = Bp[fa*ldb];
    b.y = Bp[(fa+1)*ldb];
    acc = wmma4(a, b, acc);
  }
  if (cok) {
    float bv = bias ? bias[bcol] : 0.f;
    if (bias_scalar) bv += bias_scalar[0];
    #pragma unroll
    for (int r = 0; r < 8; ++r) {
      int row = m0 + r + 8*hi;
      float v = acc[r] + bv;
      if (act == 1) v = silu_f(v);
      if (Cin) v += beta * Cin[row*ldci + bcol];
      Out[row*ldo + bcol] = v;
    }
  }
}

// ---------------- Zdiag[g,n,m,f] += H[g*64+n,f] * H[g*64+m,f]  (diagonal blocks of Z) ----------------
__global__ void k_zdiag(const float* H, float* Zd) {
  int i = blockIdx.x*256 + threadIdx.x;
  if (i >= D_B*64*64*64) return;
  int f = i & 63, m = (i >> 6) & 63, n = (i >> 12) & 63, g = i >> 18;
  Zd[i] += H[(g*64 + n)*D_F + f] * H[(g*64 + m)*D_F + f];
}

// ---------------- fused edge message kernel (16 edges / wave) ----------------
// m = silu([H[s],H[d],d2,E] @ We + be); E += m@Wme + bme (in place);
// atomics: Xnum[d] += dx*(m@Wx+bx); agg[d] += m
__global__ __launch_bounds__(32) void k_edge(
    const float* __restrict__ H, float* __restrict__ E,
    const float* __restrict__ X, const int* __restrict__ Eidx,
    const float* __restrict__ WeW, const float* __restrict__ Web,
    const float* __restrict__ WxW, const float* __restrict__ Wxb,
    const float* __restrict__ WmeW, const float* __restrict__ Wmeb,
    float* __restrict__ Xnum, float* __restrict__ agg) {
  __shared__ float As[16][164];   // K padded 161 -> 164 with zeros
  __shared__ float ms[16][64];
  __shared__ float dxs[16][4];
  int lane = threadIdx.x, l16 = lane & 15, hi = lane >> 4, koff = hi * 2;
  int e0 = blockIdx.x * 16;
  int i = lane >> 1, half = lane & 1;
  int e = e0 + i;
  int s = Eidx[e], d = Eidx[D_NE + e];
  if (half == 0) {
    float d2 = 0.f;
    #pragma unroll
    for (int c = 0; c < 3; ++c) { float dx = X[s*3+c] - X[d*3+c]; dxs[i][c] = dx; d2 += dx*dx; }
    As[i][128] = d2;
    As[i][161] = 0.f; As[i][162] = 0.f; As[i][163] = 0.f;
  }
  #pragma unroll
  for (int f = 0; f < 32; ++f) As[i][half*32 + f]        = H[s*D_F + half*32 + f];
  #pragma unroll
  for (int f = 0; f < 32; ++f) As[i][64 + half*32 + f]   = H[d*D_F + half*32 + f];
  #pragma unroll
  for (int f = 0; f < 16; ++f) As[i][129 + half*16 + f]  = E[e*D_EF + half*16 + f];
  __syncthreads();
  // m = silu(A @ We + be), A: [16 x 161(+pad)], We: [161 x 64]
  // A rows are zero for f>=161, so clamped (in-bounds) We loads contribute 0.
  #pragma unroll
  for (int nt = 0; nt < 4; ++nt) {
    v8f acc = {};
    int col = nt*16 + l16;
    #pragma unroll
    for (int kk = 0; kk < 41; ++kk) {
      int f = 4*kk + koff;
      int fb0 = (f   < 161) ? f   : 160;   // address clamp only
      int fb1 = (f+1 < 161) ? f+1 : 160;
      v2f a, b;
      a.x = As[l16][f];
      a.y = As[l16][f+1];
      b.x = WeW[fb0*D_F + col];
      b.y = WeW[fb1*D_F + col];
      acc = wmma4(a, b, acc);
    }
    float bv = Web[col];
    #pragma unroll
    for (int r = 0; r < 8; ++r) ms[r + 8*hi][col] = silu_f(acc[r] + bv);
  }
  __syncthreads();
  // coordinate numerators: Xnum[dst] += dx * (m @ Wx + bx)
  if (lane < 16) {
    float cf = Wxb[0];
    #pragma unroll 8
    for (int f = 0; f < D_F; ++f) cf += ms[lane][f] * WxW[f];
    int dd = Eidx[D_NE + e0 + lane];
    #pragma unroll
    for (int c = 0; c < 3; ++c) atomicAdd(&Xnum[dd*3 + c], dxs[lane][c] * cf);
  }
  // agg[dst] += m
  #pragma unroll 8
  for (int f = 0; f < 32; ++f) atomicAdd(&agg[d*D_F + half*32 + f], ms[i][half*32 + f]);
  __syncthreads();
  // E += m @ Wme + bme
  #pragma unroll
  for (int nt = 0; nt < 2; ++nt) {
    v8f acc = {};
    int col = nt*16 + l16;
    #pragma unroll
    for (int kk = 0; kk < 16; ++kk) {
      int f = 4*kk + koff;
      v2f a, b;
      a.x = ms[l16][f];     a.y = ms[l16][f+1];
      b.x = WmeW[f*D_EF + col];
      b.y = WmeW[(f+1)*D_EF + col];
      acc = wmma4(a, b, acc);
    }
    float bv = Wmeb[col];
    #pragma unroll
    for (int r = 0; r < 8; ++r) {
      int row = r + 8*hi;
      E[(e0 + row)*D_EF + col] += acc[r] + bv;
    }
  }
}

// ---------------- X += Xnum / max(count,1) ----------------
__global__ void k_coordupd(float* X, const float* Xnum, const float* cnt) {
  int n = blockIdx.x*256 + threadIdx.x;
  if (n >= D_N) return;
  float inv = 1.f / fmaxf(cnt[n], 1.f);
  #pragma unroll
  for (int c = 0; c < 3; ++c) X[n*3 + c] += Xnum[n*3 + c] * inv;
}

// ---------------- HA = [H | agg] ----------------
__global__ void k_buildha(const float* H, const float* agg, float* HA) {
  int i = blockIdx.x*256 + threadIdx.x;
  if (i >= D_N*128) return;
  int n = i >> 7, c = i & 127;
  HA[i] = (c < 64) ? H[n*D_F + c] : agg[n*D_F + (c - 64)];
}

// ---------------- pair-biased attention, one wave per (graph, head) ----------------
__global__ __launch_bounds__(32) void k_attn(
    const float* __restrict__ q, const float* __restrict__ k,
    const float* __restrict__ v, const float* __restrict__ BiasL,
    float* __restrict__ o) {
  __shared__ float sc[64][64];
  int lane = threadIdx.x, l16 = lane & 15, hi = lane >> 4, koff = hi * 2;
  int g = blockIdx.x / D_NH, h = blockIdx.x % D_NH;
  const float scale = 0.25f; // 1/sqrt(HD=16)
  // scores = q k^T * scale + bias   (64x64, fully in-graph: mask is block-diagonal)
  #pragma unroll
  for (int tn = 0; tn < 4; ++tn)
    for (int tm = 0; tm < 4; ++tm) {
      v8f acc = {};
      #pragma unroll
      for (int kk = 0; kk < 4; ++kk) {
        int f = 4*kk + koff;
        v2f a, b;
        a.x = q[(g*64 + tn*16 + l16)*D_IN + h*D_HD + f];
        a.y = q[(g*64 + tn*16 + l16)*D_IN + h*D_HD + f + 1];
        b.x = k[(g*64 + tm*16 + l16)*D_IN + h*D_HD + f];
        b.y = k[(g*64 + tm*16 + l16)*D_IN + h*D_HD + f + 1];
        acc = wmma4(a, b, acc);
      }
      #pragma unroll
      for (int r = 0; r < 8; ++r) {
        int rr = tn*16 + r + 8*hi, cc = tm*16 + l16;
        sc[rr][cc] = acc[r]*scale + BiasL[(g*4096 + rr*64 + cc)*D_NH + h];
      }
    }
  __syncthreads();
  // row softmax over 64 in-graph columns
  for (int rr = lane; rr < 64; rr += 32) {
    float mx = -1e30f;
    #pragma unroll 8
    for (int j = 0; j < 64; ++j) mx = fmaxf(mx, sc[rr][j]);
    float sum = 0.f;
    #pragma unroll 8
    for (int j = 0; j < 64; ++j) { float p = __expf(sc[rr][j] - mx); sc[rr][j] = p; sum += p; }
    float inv = 1.f / sum;
    #pragma unroll 8
    for (int j = 0; j < 64; ++j) sc[rr][j] *= inv;
  }
  __syncthreads();
  // o = P @ V  (64x64 @ 64x16)
  #pragma unroll
  for (int tn = 0; tn < 4; ++tn) {
    v8f acc = {};
    #pragma unroll
    for (int kk = 0; kk < 16; ++kk) {
      int f = 4*kk + koff;
      v2f a, b;
      a.x = sc[tn*16 + l16][f];
      a.y = sc[tn*16 + l16][f+1];
      b.x = v[(g*64 + f)*D_IN + h*D_HD + l16];
      b.y = v[(g*64 + f + 1)*D_IN + h*D_HD + l16];
      acc = wmma4(a, b, acc);
    }
    #pragma unroll
    for (int r = 0; r < 8; ++r)
      o[(g*64 + tn*16 + r + 8*hi)*D_IN + h*D_HD + l16] = acc[r];
  }
}

// ---------------- recenter coordinates per graph ----------------
__global__ void k_recenter(float* X) {
  __shared__ float sx[64], sy[64], sz[64];
  int g = blockIdx.x, t = threadIdx.x, n = g*64 + t;
  float x = X[n*3], y = X[n*3+1], z = X[n*3+2];
  sx[t] = x; sy[t] = y; sz[t] = z;
  __syncthreads();
  for (int s2 = 32; s2 > 0; s2 >>= 1) {
    if (t < s2) { sx[t]+=sx[t+s2]; sy[t]+=sy[t+s2]; sz[t]+=sz[t+s2]; }
    __syncthreads();
  }
  X[n*3]   = x - sx[0]/64.f;
  X[n*3+1] = y - sy[0]/64.f;
  X[n*3+2] = z - sz[0]/64.f;
}

// ---------------- build A (scaled checkpoints) and B (transposed stack) for z_logits ----------------
__global__ void k_prepz(const float* Hck, const float* hz, float* Ahz, float* Bt) {
  int i = blockIdx.x*256 + threadIdx.x;
  if (i >= D_N * 384) return;
  int n = i / 384, r = i % 384, c = r / 64, f = r % 64;
  float val = Hck[c*D_N*D_F + n*D_F + f];
  Ahz[n*384 + r] = val * hz[f];
  Bt[r*D_N + n]  = val;
}

// ---------------- host launcher ----------------
extern "C" void kernel_launch(void* const* d_in, const int* in_sizes, int n_in,
                              void* d_out, int out_size, void* d_ws, size_t ws_size,
                              hipStream_t stream) {
  (void)in_sizes; (void)n_in; (void)out_size; (void)ws_size;
  auto F32 = [&](int i){ return (const float*)d_in[i]; };
  auto I32 = [&](int i){ return (const int*)d_in[i]; };
  // d_in layout (setup_inputs order, params tree-flattened with sorted dict keys):
  // 0:X 1:H 2:E 3:t 4:batch 5:E_idx
  // 6,7:ada_scale 8,9:ada_shift 10,11:atom_emb 12,13:edge_emb
  // 14,15:head_e 16,17:head_h 18,19:head_z
  // 20+17*l: layer l = We(W,b) Wh Wk Wme Wo Wq Wv Wx (W,b each) Wzb
  // 105,106:time1 107,108:time2
  float* ws = (float*)d_ws;
  float* H       = ws + 0;        // 49152
  float* E       = ws + 49152;    // 786432
  float* X       = ws + 835584;   // 2304
  float* scale_g = ws + 837888;   // 768
  float* shift_g = ws + 838656;   // 768
  float* Zdiag   = ws + 839424;   // 3145728
  float* BiasL   = ws + 3985152;  // 491520
  float* Hck     = ws + 4476672;  // 294912 (6 checkpoints)
  float* q       = ws + 4771584;  // 122880
  float* k       = ws + 4894464;
  float* v       = ws + 5017344;
  float* o       = ws + 5140224;
  float* HA      = ws + 5263104;  // 98304
  float* agg     = ws + 5361408;  // 49152
  float* Xnum    = ws + 5410560;  // 2304
  float* cnt     = ws + 5412864;  // 768
  float* Ahz     = ws + 5413632;  // 294912
  float* Bt      = ws + 5708544;  // 294912  (total ~24 MB)
  float* outX = (float*)d_out;
  float* outH = outX + 2304;
  float* outE = outX + 14592;
  float* outZ = outX + 137472;

  dim3 b256(256), b32(32), b64(64);
  #define GRD(n) dim3(((n) + 255) / 256)

  // setup
  k_zero<<<GRD(D_B*64*64*64), b256, 0, stream>>>(Zdiag, D_B*64*64*64);
  k_zero<<<GRD(D_N), b256, 0, stream>>>(cnt, D_N);
  k_copy<<<GRD(D_N*3), b256, 0, stream>>>(X, F32(0), D_N*3);
  k_time<<<D_B, b64, 0, stream>>>(F32(3), F32(105), F32(106), F32(107), F32(108),
                                  F32(6), F32(7), F32(8), F32(9), scale_g, shift_g);
  k_embed_e<<<GRD(D_NE*D_EF), b256, 0, stream>>>(F32(2), F32(12), F32(13), E);
  k_embed_h<<<D_N, b64, 0, stream>>>(F32(1), F32(10), F32(11), I32(4), scale_g, shift_g, H);
  k_counts<<<GRD(D_NE), b256, 0, stream>>>(I32(5), cnt);
  k_copy<<<GRD(D_N*D_F), b256, 0, stream>>>(Hck, H, D_N*D_F);        // checkpoint 0
  k_zdiag<<<GRD(D_B*64*64*64), b256, 0, stream>>>(H, Zdiag);

  for (int l = 0; l < D_NL; ++l) {
    int base = 20 + 17*l;
    const float *WeW = F32(base+0),  *Web  = F32(base+1);
    const float *WhW = F32(base+2),  *Whb  = F32(base+3);
    const float *WkW = F32(base+4),  *Wkb  = F32(base+5);
    const float *WmeW= F32(base+6),  *Wmeb = F32(base+7);
    const float *WoW = F32(base+8),  *Wob  = F32(base+9);
    const float *WqW = F32(base+10), *Wqb  = F32(base+11);
    const float *WvW = F32(base+12), *Wvb  = F32(base+13);
    const float *WxW = F32(base+14), *Wxb  = F32(base+15);
    const float *Wzb = F32(base+16);

    k_zero<<<GRD(D_N*D_F), b256, 0, stream>>>(agg, D_N*D_F);
    k_zero<<<GRD(D_N*3), b256, 0, stream>>>(Xnum, D_N*3);
    k_edge<<<D_NE/16, b32, 0, stream>>>(H, E, X, I32(5), WeW, Web, WxW, Wxb, WmeW, Wmeb, Xnum, agg);
    k_coordupd<<<GRD(D_N), b256, 0, stream>>>(X, Xnum, cnt);
    k_buildha<<<GRD(D_N*128), b256, 0, stream>>>(H, agg, HA);
    // H = H + silu(HA @ Wh + b)
    k_gemm<<<dim3((D_N/16)*4), b32, 0, stream>>>(HA, 128, WhW, 64, Whb, nullptr, H, 64, H, 64,
                                                 64, 128, 4, 1, 1.f);
    // q,k,v projections
    k_gemm<<<dim3((D_N/16)*10), b32, 0, stream>>>(H, 64, WqW, D_IN, Wqb, nullptr, nullptr, 0, q, D_IN,
                                                  D_IN, 64, 10, 0, 0.f);
    k_gemm<<<dim3((D_N/16)*10), b32, 0, stream>>>(H, 64, WkW, D_IN, Wkb, nullptr, nullptr, 0, k, D_IN,
                                                  D_IN, 64, 10, 0, 0.f);
    k_gemm<<<dim3((D_N/16)*10), b32, 0, stream>>>(H, 64, WvW, D_IN, Wvb, nullptr, nullptr, 0, v, D_IN,
                                                  D_IN, 64, 10, 0, 0.f);
    // per-layer pair bias on diagonal blocks: BiasL[(g,n,m),h] = Zdiag @ Wzb_l
    k_gemm<<<dim3((D_B*64*64/16)*1), b32, 0, stream>>>(Zdiag, 64, Wzb, D_NH, nullptr, nullptr,
                                                       nullptr, 0, BiasL, D_NH,
                                                       D_NH, 64, 1, 0, 0.f);
    k_attn<<<D_B*D_NH, b32, 0, stream>>>(q, k, v, BiasL, o);
    // H = H + o @ Wo + b
    k_gemm<<<dim3((D_N/16)*4), b32, 0, stream>>>(o, D_IN, WoW, 64, Wob, nullptr, H, 64, H, 64,
                                                 64, D_IN, 4, 0, 1.f);
    // checkpoint l+1, accumulate Z diagonal (only needed for bias if more layers remain)
    k_copy<<<GRD(D_N*D_F), b256, 0, stream>>>(Hck + (l+1)*D_N*D_F, H, D_N*D_F);
    if (l < D_NL - 1)
      k_zdiag<<<GRD(D_B*64*64*64), b256, 0, stream>>>(H, Zdiag);
    k_recenter<<<D_B, b64, 0, stream>>>(X);
  }

  // outputs
  k_copy<<<GRD(D_N*3), b256, 0, stream>>>(outX, X, D_N*3);
  k_gemm<<<dim3((D_N/16)*1), b32, 0, stream>>>(H, 64, F32(16), 16, F32(17), nullptr, nullptr, 0,
                                               outH, 16, 16, 64, 1, 0, 0.f);
  k_gemm<<<dim3((D_NE/16)*1), b32, 0, stream>>>(E, D_EF, F32(14), D_EC, F32(15), nullptr, nullptr, 0,
                                                outE, D_EC, D_EC, D_EF, 1, 0, 0.f);
  // z_logits = sum_c (Hc * hz) @ Hc^T + b  (symmetric by construction)
  k_prepz<<<GRD(D_N*384), b256, 0, stream>>>(Hck, F32(18), Ahz, Bt);
  k_gemm<<<dim3((D_N/16)*(D_N/16)), b32, 0, stream>>>(Ahz, 384, Bt, D_N, nullptr, F32(19),
                                                      nullptr, 0, outZ, D_N,
                                                      D_N, 384, D_N/16, 0, 0.f);
  #undef GRD
}